// DynamicDifferentiablePalette_14482629722924
// MI455X (gfx1250) — compile-verified
//
#include <hip/hip_runtime.h>
#include <hip/hip_bf16.h>

// ---------------------------------------------------------------------------
// DynamicDifferentiablePalette on MI455X (gfx1250, wave32)
//
// logits_k = (2*x . p_k - |p_k|^2) * (log2e/T)   -- softmax-equivalent to
// -|x-p_k|^2/T (row-constant shift cancels; log2e pre-scale lets us use the
// native v_exp_f32 (= exp2) with no per-element multiply).
//
// V_WMMA_F32_16X16X4_F32 with A = palette_aug (16 entries x 4),
// B = pixel_aug^T (4 x 16 pixels)  ->  D[k_row, pixel_lane].
// Each lane owns 32 of its pixel's 64 logits in registers: softmax and blend
// reduce in-register + one v_permlanex16 half-swap per quantity. Blend uses
// packed f32 FMAs (V_PK_FMA_F32); denominator rides in the .w slot (p.w = 1).
//
// Ragged mask is free: padded palette entries get K3 = -3e38 at stage time,
// so their WMMA logits are ~-3e38 and exp2 underflows to exactly 0 -- no
// per-element cndmask needed (all intermediates stay finite; ksize >= 1).
// ---------------------------------------------------------------------------

typedef __attribute__((ext_vector_type(2))) float v2f;
typedef __attribute__((ext_vector_type(8))) float v8f;

#define BATCH    16
#define HEIGHT   256
#define WIDTH    256
#define CHANNELS 3
#define KMAX     64

#define LOG2E 1.4426950408889634f

// lane i <-> lane i^16 exchange (identity nibble pattern, cross-row select)
__device__ __forceinline__ float swap_half(float v) {
    int s = __builtin_bit_cast(int, v);
    int r = __builtin_amdgcn_permlanex16(s, s, 0x76543210, (int)0xFEDCBA98,
                                         false, false);
    return __builtin_bit_cast(float, r);
}

__global__ __launch_bounds__(256) void palette_wmma_kernel(
    const float* __restrict__ images,          // [B,H,W,3]
    const float* __restrict__ palettes,        // [B,64,3]
    const int*   __restrict__ palette_sizes,   // [B]
    const float* __restrict__ temperature,     // scalar
    float*       __restrict__ out)             // [B,H,W,3]
{
    __shared__ float4 palScaled[KMAX];  // (2p0, 2p1, 2p2, -|p|^2)*log2e/T,
                                        // K3 poisoned to -3e38 for k >= ksize
    __shared__ float4 palRaw[KMAX];     // {p0, p1, p2, 1.0}

    const int b   = blockIdx.x / HEIGHT;
    const int h   = blockIdx.x % HEIGHT;
    const int tid = threadIdx.x;

    // --- Stage palette tables into LDS (once per block) --------------------
    if (tid < KMAX) {
        const float s = LOG2E / temperature[0];
        const int   ksize = palette_sizes[b];
        const float* p = palettes + ((size_t)b * KMAX + tid) * CHANNELS;
        const float p0 = p[0], p1 = p[1], p2 = p[2];
        // Padded slots: poisoned constant term -> logit ~ -3e38 -> exp2 -> 0.
        const float k3 = (tid < ksize)
                             ? -(p0 * p0 + p1 * p1 + p2 * p2) * s
                             : -3.0e38f;
        palScaled[tid] = make_float4(2.0f * p0 * s,
                                     2.0f * p1 * s,
                                     2.0f * p2 * s,
                                     k3);
        palRaw[tid] = make_float4(p0, p1, p2, 1.0f);  // .w=1 -> sum(e) for free
    }
    __syncthreads();

    const int wave  = tid >> 5;        // 8 waves per block
    const int lane  = tid & 31;
    const int l16   = lane & 15;       // pixel index within tile (N lanes)
    const int hi    = lane >> 4;       // half select: K2/K3 ops, rows k+8
    const int hi8   = hi << 3;

    const size_t rowBase = (((size_t)b * HEIGHT + h) * WIDTH) * CHANNELS;

    // Warm the wave's second tile (global_prefetch_b8).
    __builtin_prefetch(images + rowBase + (size_t)((wave * 2 + 1) * 16 + l16) * CHANNELS, 0, 1);

    // A-fragments: palette chunks, loop-invariant (4 x ds_load_b64 total).
    // A layout (16x4 f32): lanes 0-15 hold {2p0*s, 2p1*s}, lanes 16-31 hold
    // {2p2*s, K3} for entry m = i*16 + l16.
    v2f apal[4];
#pragma unroll
    for (int i = 0; i < 4; ++i) {
        const float* ps = (const float*)&palScaled[i * 16 + l16];
        apal[i].x = ps[hi * 2 + 0];
        apal[i].y = ps[hi * 2 + 1];
    }

    // --- Each wave: 2 tiles of 16 pixels along this image row --------------
    for (int t = wave * 2; t < wave * 2 + 2; ++t) {
        const int w0 = t * 16;
        const float* pix = images + rowBase + (size_t)(w0 + l16) * CHANNELS;

        // B-fragment (4x16): lanes 0-15 -> {x0, x1}, lanes 16-31 -> {x2, 1}.
        // Unconditional loads (no exec branching): pix[1] is always in-bounds.
        const float f0 = __builtin_nontemporal_load(pix + hi * 2);
        const float f1 = __builtin_nontemporal_load(pix + 1);
        v2f bpix;
        bpix.x = f0;
        bpix.y = hi ? 1.0f : f1;

        // 4 WMMAs: D[i] rows r -> palette k = i*16 + hi8 + r, lanes -> pixel.
        v8f d[4];
#pragma unroll
        for (int i = 0; i < 4; ++i) {
            v8f c = {};
            d[i] = __builtin_amdgcn_wmma_f32_16x16x4_f32(
                /*neg_a=*/false, apal[i], /*neg_b=*/false, bpix,
                /*c_mod=*/(short)0, c, /*reuse_a=*/false, /*reuse_b=*/false);
        }

        // --- Softmax max: in-register tree over 32 logits, then half-swap.
        // (Poisoned padded logits are ~-3e38: never the max since ksize >= 1.)
        float mx = d[0][0];
#pragma unroll
        for (int i = 0; i < 4; ++i)
#pragma unroll
            for (int r = 0; r < 8; ++r)
                mx = fmaxf(mx, d[i][r]);
        mx = fmaxf(mx, swap_half(mx));      // full 64-entry max per pixel

        // --- exp2 + packed blend accumulation (mask-free) ------------------
        v2f axy = {0.0f, 0.0f};
        v2f azw = {0.0f, 0.0f};             // .y accumulates sum(e)
#pragma unroll
        for (int i = 0; i < 4; ++i) {
#pragma unroll
            for (int r = 0; r < 8; ++r) {
                const int k = i * 16 + r + hi8;             // this half's k
                const float e = __builtin_amdgcn_exp2f(d[i][r] - mx); // ==0 if padded
                const float4 p = palRaw[k];                 // LDS broadcast
                v2f ee;  ee.x  = e;   ee.y  = e;
                v2f pxy; pxy.x = p.x; pxy.y = p.y;
                v2f pzw; pzw.x = p.z; pzw.y = p.w;          // p.w == 1.0
                axy += ee * pxy;                            // v_pk_fma_f32
                azw += ee * pzw;
            }
        }

        // Combine the two lane-halves (k = i*16 + r | k = i*16 + 8 + r per half).
        const float ax = axy.x + swap_half(axy.x);
        const float ay = axy.y + swap_half(axy.y);
        const float az = azw.x + swap_half(azw.x);
        const float aw = azw.y + swap_half(azw.y);

        const float rcp = __builtin_amdgcn_rcpf(aw);

        // Coalesced store: lane = pixel. Lo half writes c0,c1; hi half c2.
        float* o = out + rowBase + (size_t)(w0 + l16) * CHANNELS;
        if (hi == 0) {
            __builtin_nontemporal_store(ax * rcp, o + 0);
            __builtin_nontemporal_store(ay * rcp, o + 1);
        } else {
            __builtin_nontemporal_store(az * rcp, o + 2);
        }
    }
}

extern "C" void kernel_launch(void* const* d_in, const int* in_sizes, int n_in,
                              void* d_out, int out_size, void* d_ws, size_t ws_size,
                              hipStream_t stream) {
    const float* images        = (const float*)d_in[0];
    const float* palettes      = (const float*)d_in[1];
    const int*   palette_sizes = (const int*)d_in[2];
    const float* temperature   = (const float*)d_in[3];
    float*       out           = (float*)d_out;

    (void)in_sizes; (void)n_in; (void)out_size; (void)d_ws; (void)ws_size;

    dim3 grid(BATCH * HEIGHT);   // one block per image row
    dim3 block(256);             // 8 wave32s, 2 pixel tiles each
    palette_wmma_kernel<<<grid, block, 0, stream>>>(
        images, palettes, palette_sizes, temperature, out);
}